// PostProcessor_9045201125727
// MI455X (gfx1250) — compile-verified
//
#include <hip/hip_runtime.h>
#include <hip/hip_bf16.h>
#include <stdint.h>

// Problem constants (from reference)
#define BB      2
#define NP      2048          // proposals per image
#define NCLS    81
#define NFG     80            // foreground classes
#define DETS    100
#define THREADS 256
#define SCORE_THRESH 0.05f
#define NMS_THR      0.5f
#define NEG_INF     (-1000000000.0f)
#define DW_CLAMP     4.135166556742356f   // log(1000/16)

// ---------------------------------------------------------------------------
// CDNA5 async global->LDS DMA helpers (ASYNCcnt path, gfx1250-specific).
// LDS byte address from the vdst VGPR;  global = SGPR64 + VGPR32 offset (GVS mode).
// Low 32 bits of a generic shared pointer == LDS byte offset on gfx1250.
// ---------------------------------------------------------------------------
__device__ __forceinline__ void async_load_b128(void* lds_ptr, unsigned byte_off,
                                                const void* sbase) {
  unsigned lds_addr = (unsigned)(uintptr_t)lds_ptr;
  uint64_t base = (uint64_t)(uintptr_t)sbase;
  asm volatile("global_load_async_to_lds_b128 %0, %1, %2"
               :: "v"(lds_addr), "v"(byte_off), "s"(base) : "memory");
}
__device__ __forceinline__ void wait_async0() {
  asm volatile("s_wait_asynccnt 0" ::: "memory");
}

// ---------------------------------------------------------------------------
// Kernel A: softmax + per-class box decode/clip, transposed to [b][cls][n].
// Exact-rounded divisions (not reciprocal multiplies) to track the reference
// tightly near the score/IoU decision thresholds.
// ---------------------------------------------------------------------------
__global__ void decode_kernel(const float* __restrict__ logits,
                              const float* __restrict__ regr,
                              const float* __restrict__ props,
                              const int* __restrict__ im_w_p,
                              const int* __restrict__ im_h_p,
                              float* __restrict__ scores_t,
                              float4* __restrict__ boxes_t) {
  int gid = blockIdx.x * blockDim.x + threadIdx.x;
  if (gid >= BB * NP) return;
  int b = gid / NP, n = gid % NP;

  const float* lg = logits + (size_t)gid * NCLS;
  float m = lg[0];
  for (int c = 1; c < NCLS; ++c) m = fmaxf(m, lg[c]);
  float sum = 0.f;
  for (int c = 0; c < NCLS; ++c) sum += expf(lg[c] - m);

  const float* pb = props + (size_t)gid * 4;
  float w  = pb[2] - pb[0] + 1.f;
  float h  = pb[3] - pb[1] + 1.f;
  float cx = pb[0] + 0.5f * w;
  float cy = pb[1] + 0.5f * h;
  float W1 = (float)(*im_w_p) - 1.f;
  float H1 = (float)(*im_h_p) - 1.f;

  const float* rg = regr + (size_t)gid * 4 * NCLS;
  for (int j = 1; j < NCLS; ++j) {
    float dx = rg[4*j+0] / 10.f;
    float dy = rg[4*j+1] / 10.f;
    float dw = fminf(rg[4*j+2] / 5.f, DW_CLAMP);
    float dh = fminf(rg[4*j+3] / 5.f, DW_CLAMP);
    float pcx = dx * w + cx;
    float pcy = dy * h + cy;
    float pw  = expf(dw) * w;
    float ph  = expf(dh) * h;
    float x1 = fminf(fmaxf(pcx - 0.5f * pw, 0.f), W1);
    float y1 = fminf(fmaxf(pcy - 0.5f * ph, 0.f), H1);
    float x2 = fminf(fmaxf(pcx + 0.5f * pw - 1.f, 0.f), W1);
    float y2 = fminf(fmaxf(pcy + 0.5f * ph - 1.f, 0.f), H1);
    size_t o = ((size_t)b * NFG + (j - 1)) * NP + n;
    scores_t[o] = expf(lg[j] - m) / sum;
    boxes_t[o]  = make_float4(x1, y1, x2, y2);
  }
}

// ---------------------------------------------------------------------------
// Kernel B: one workgroup per (image, class).  Async-stage scores to LDS,
// bitonic sort (stable desc), async-gather sorted boxes, tiled bitmask NMS
// with wave32 ballot resolution.  Whole per-class problem lives in ~64KB LDS
// (320KB/WGP on CDNA5).
// ---------------------------------------------------------------------------
__global__ void __launch_bounds__(THREADS)
nms_kernel(const float* __restrict__ scores_t,
           const float4* __restrict__ boxes_t,
           float* __restrict__ scores_s,
           float4* __restrict__ boxes_s) {
  __shared__ float    s_key[NP];        // masked scores (sorted)
  __shared__ int      s_idx[NP];        // original proposal index
  __shared__ float4   s_box[NP];        // sorted decoded boxes
  __shared__ float    s_area[NP];
  __shared__ unsigned s_keep[NP / 32];  // kept bitmask
  __shared__ unsigned s_sup[32 * 64];   // 32-row x 2048-col suppression tile

  const int tid = threadIdx.x;
  const int b = blockIdx.x / NFG;
  const int j = blockIdx.x % NFG;
  const size_t row = ((size_t)b * NFG + j) * NP;
  const float*  srow = scores_t + row;
  const float4* brow = boxes_t + row;

  // --- stage scores into LDS via async DMA (2 x b128 per thread = 8 KB) ----
  for (int r = 0; r < 2; ++r) {
    int grp = r * THREADS + tid;                 // group of 4 floats
    async_load_b128(&s_key[grp * 4], (unsigned)(grp * 16), srow);
  }
  wait_async0();
  __syncthreads();

  // mask by score threshold, init indices / keep mask
  for (int i = tid; i < NP; i += THREADS) {
    float sc = s_key[i];
    s_key[i] = (sc > SCORE_THRESH) ? sc : NEG_INF;
    s_idx[i] = i;
  }
  if (tid < NP / 32) s_keep[tid] = 0u;
  __syncthreads();

  // --- bitonic sort: descending by key, ties -> ascending index (stable) ---
  for (int k = 2; k <= NP; k <<= 1) {
    for (int mgap = k >> 1; mgap > 0; mgap >>= 1) {
      for (int i = tid; i < NP; i += THREADS) {
        int ixj = i ^ mgap;
        if (ixj > i) {
          float ka = s_key[i], kb = s_key[ixj];
          int   ia = s_idx[i], ib = s_idx[ixj];
          bool a_first = (ka > kb) || (ka == kb && ia < ib); // strict total order
          bool desc = ((i & k) == 0);
          if (desc ? !a_first : a_first) {
            s_key[i] = kb; s_key[ixj] = ka;
            s_idx[i] = ib; s_idx[ixj] = ia;
          }
        }
      }
      __syncthreads();
    }
  }

  // --- gather sorted boxes: per-lane indexed async DMA (global -> LDS) -----
  for (int i = tid; i < NP; i += THREADS) {
    async_load_b128(&s_box[i], (unsigned)(s_idx[i] * 16), brow);
  }
  wait_async0();
  __syncthreads();

  for (int i = tid; i < NP; i += THREADS) {
    float4 bx = s_box[i];
    s_area[i] = (bx.z - bx.x + 1.f) * (bx.w - bx.y + 1.f);
  }
  __syncthreads();

  // --- tiled greedy NMS ----------------------------------------------------
  for (int t0 = 0; t0 < NP; t0 += 32) {
    // parallel phase: suppression bits for 32 rows vs all 2048 columns
    for (int wi = tid; wi < 32 * 64; wi += THREADS) {
      int r = wi >> 6;          // row within tile
      int w = wi & 63;          // 32-column word
      float4 a = s_box[t0 + r];
      float areaA = s_area[t0 + r];
      unsigned bits = 0u;
      int jbase = w * 32;
      for (int uu = 0; uu < 32; ++uu) {
        int u = (uu + w) & 31;  // rotate start to spread LDS banks
        int jj = jbase + u;
        float4 bx = s_box[jj];
        float iw = fmaxf(fminf(a.z, bx.z) - fmaxf(a.x, bx.x) + 1.f, 0.f);
        float ih = fmaxf(fminf(a.w, bx.w) - fmaxf(a.y, bx.y) + 1.f, 0.f);
        float inter = iw * ih;
        float iou = inter / (areaA + s_area[jj] - inter);
        if (iou > NMS_THR) bits |= (1u << u);
      }
      s_sup[r * 64 + w] = bits;
    }
    __syncthreads();

    // serial phase: wave 0 resolves 32 decisions wave-synchronously
    if (tid < 32) {
      volatile unsigned* keep = s_keep;
      for (int r = 0; r < 32; ++r) {
        int i = t0 + r;
        unsigned c0 = s_sup[r * 64 + tid * 2]     & keep[tid * 2];
        unsigned c1 = s_sup[r * 64 + tid * 2 + 1] & keep[tid * 2 + 1];
        unsigned long long any = __ballot((c0 | c1) != 0u);
        bool keep_i = (s_key[i] > SCORE_THRESH) && (any == 0ull);
        if (keep_i && tid == 0) keep[i >> 5] |= (1u << (i & 31));
        __builtin_amdgcn_wave_barrier();   // order LDS write before next reads
      }
    }
    __syncthreads();
  }

  // --- emit sorted boxes + kept-masked scores ------------------------------
  for (int i = tid; i < NP; i += THREADS) {
    bool kept = (s_keep[i >> 5] >> (i & 31)) & 1u;
    scores_s[row + i] = kept ? s_key[i] : NEG_INF;
    boxes_s[row + i]  = s_box[i];
  }
}

// ---------------------------------------------------------------------------
// Kernel C: per-image top-100 (100 argmax passes, tie -> lowest flat index).
// Selected-mask kept in LDS so d_ws is never mutated (deterministic replays).
// ---------------------------------------------------------------------------
#define M (NFG * NP)   // 163840 candidates per image
__global__ void __launch_bounds__(THREADS)
topk_kernel(const float* __restrict__ scores_s,
            const float4* __restrict__ boxes_s,
            float* __restrict__ out) {
  __shared__ unsigned selmask[M / 32];   // 20 KB
  __shared__ float red_v[THREADS];
  __shared__ int   red_i[THREADS];

  const int tid = threadIdx.x;
  const int b = blockIdx.x;
  const float*  sc = scores_s + (size_t)b * M;
  const float4* bx = boxes_s + (size_t)b * M;

  for (int i = tid; i < M / 32; i += THREADS) selmask[i] = 0u;
  __syncthreads();

  for (int k = 0; k < DETS; ++k) {
    float best = -INFINITY;
    int   bi = M;
    for (int i = tid; i < M; i += THREADS) {
      bool taken = (selmask[i >> 5] >> (i & 31)) & 1u;
      float v = taken ? -INFINITY : sc[i];
      if (v > best || (v == best && i < bi)) { best = v; bi = i; }
    }
    red_v[tid] = best; red_i[tid] = bi;
    __syncthreads();
    for (int s = THREADS / 2; s > 0; s >>= 1) {
      if (tid < s) {
        float v2 = red_v[tid + s]; int i2 = red_i[tid + s];
        if (v2 > red_v[tid] || (v2 == red_v[tid] && i2 < red_i[tid])) {
          red_v[tid] = v2; red_i[tid] = i2;
        }
      }
      __syncthreads();
    }
    if (tid == 0) {
      int sel = red_i[0];
      float4 bbx = bx[sel];
      float* ob = out + ((size_t)b * DETS + k) * 4;        // boxes [0,800)
      ob[0] = bbx.x; ob[1] = bbx.y; ob[2] = bbx.z; ob[3] = bbx.w;
      out[BB * DETS * 4 + b * DETS + k] = red_v[0];        // scores [800,1000)
      out[BB * DETS * 5 + b * DETS + k] = (float)(sel / NP + 1); // labels
      selmask[sel >> 5] |= (1u << (sel & 31));
    }
    __syncthreads();
  }
}

// ---------------------------------------------------------------------------
extern "C" void kernel_launch(void* const* d_in, const int* in_sizes, int n_in,
                              void* d_out, int out_size, void* d_ws, size_t ws_size,
                              hipStream_t stream) {
  const float* logits = (const float*)d_in[0];
  const float* regr   = (const float*)d_in[1];
  const float* props  = (const float*)d_in[2];
  const int*   im_w   = (const int*)d_in[3];
  const int*   im_h   = (const int*)d_in[4];

  // workspace layout (floats), all float4 regions 16B-aligned (~12.5 MB total)
  float*  ws       = (float*)d_ws;
  float*  scores_t = ws;                                       // B*80*N
  float4* boxes_t  = (float4*)(scores_t + (size_t)BB*NFG*NP);  // B*80*N float4
  float*  scores_s = (float*)(boxes_t + (size_t)BB*NFG*NP);    // B*80*N
  float4* boxes_s  = (float4*)(scores_s + (size_t)BB*NFG*NP);  // B*80*N float4

  decode_kernel<<<(BB * NP + THREADS - 1) / THREADS, THREADS, 0, stream>>>(
      logits, regr, props, im_w, im_h, scores_t, boxes_t);
  nms_kernel<<<BB * NFG, THREADS, 0, stream>>>(
      scores_t, boxes_t, scores_s, boxes_s);
  topk_kernel<<<BB, THREADS, 0, stream>>>(
      scores_s, boxes_s, (float*)d_out);
}